// BaseRouter_3435973837290
// MI455X (gfx1250) — compile-verified
//
#include <hip/hip_runtime.h>
#include <math.h>

typedef __attribute__((ext_vector_type(2))) float v2f;
typedef __attribute__((ext_vector_type(4))) float v4f;
typedef __attribute__((ext_vector_type(8))) float v8f;

#define HIDDEN 1024
#define NEXP   8
#define NTOK   4096              // B*S = 2*2048
#define CAP    1536              // int(2*2048*1.5*2/8)

#define DISP_ELEMS (NTOK * NEXP * CAP)          // 50331648 floats
#define PROBS_OFF  (2 * DISP_ELEMS)             // 100663296
#define AUX_OFF    (PROBS_OFF + NTOK * NEXP)    // 100696064

// ---------------------------------------------------------------------------
// Kernel 1: zero the dispatch/combine region (402 MB) with non-temporal b128
// stores (streaming: don't evict X/W1/h from L2), and zero the 8 per-expert
// probability accumulators in workspace.
// ---------------------------------------------------------------------------
__global__ __launch_bounds__(256) void zero_fill_kernel(v4f* __restrict__ out,
                                                        long n4,
                                                        float* __restrict__ esum) {
  long i = (long)blockIdx.x * blockDim.x + threadIdx.x;
  long stride = (long)gridDim.x * blockDim.x;
  v4f z = {};
  for (; i < n4; i += stride) __builtin_nontemporal_store(z, &out[i]);
  if (blockIdx.x == 0 && threadIdx.x < NEXP) esum[threadIdx.x] = 0.0f;
}

// ---------------------------------------------------------------------------
// Kernel 2: h = relu(X @ W1^T + b1) via V_WMMA_F32_16X16X4_F32.
// Register-blocked: one wave computes a 4x2 grid of 16x16 tiles (64 rows x
// 32 cols of h) -> 8 independent accumulator chains (pipelined WMMAs), A
// fragments reused x2 and B fragments x4 (6 loads feed 8 WMMAs per K-step).
//   M blocks: 4096/64 = 64, N blocks: 1024/32 = 32 -> 2048 waves, 4/block.
// A (16x4, MxK): lane 0-15 -> M=lane, VGPR0/1 = K0/K1; lane 16-31 -> K2/K3.
// B (4x16, KxN): lane&15 -> N col, VGPR0/1 = K rows (khalf, khalf+1).
// C/D (16x16): col = lane&15, row = v + 8*(lane>>4) for VGPR v in 0..7.
// ---------------------------------------------------------------------------
__global__ __launch_bounds__(128) void gemm16_relu_kernel(
    const float* __restrict__ X,    // [4096,1024]
    const float* __restrict__ W1,   // [1024,1024]  (row o holds weights over h)
    const float* __restrict__ B1,   // [1024]
    float* __restrict__ H)          // [4096,1024]
{
  const int wave = threadIdx.x >> 5;
  const int lane = threadIdx.x & 31;
  const int wid  = blockIdx.x * 4 + wave;     // 0..2047
  const int tm   = wid >> 5;                  // 0..63   (M block: 64 rows)
  const int tn   = wid & 31;                  // 0..31   (N block: 32 cols)

  const int lm    = lane & 15;
  const int khalf = (lane >> 4) << 1;         // 0 or 2

  const float* __restrict__ arow0 = X  + (size_t)(tm * 64 + 0 * 16 + lm) * HIDDEN + khalf;
  const float* __restrict__ arow1 = X  + (size_t)(tm * 64 + 1 * 16 + lm) * HIDDEN + khalf;
  const float* __restrict__ arow2 = X  + (size_t)(tm * 64 + 2 * 16 + lm) * HIDDEN + khalf;
  const float* __restrict__ arow3 = X  + (size_t)(tm * 64 + 3 * 16 + lm) * HIDDEN + khalf;
  const float* __restrict__ brow0 = W1 + (size_t)(tn * 32 + 0 * 16 + lm) * HIDDEN + khalf;
  const float* __restrict__ brow1 = W1 + (size_t)(tn * 32 + 1 * 16 + lm) * HIDDEN + khalf;

  v8f acc00 = {}, acc01 = {};
  v8f acc10 = {}, acc11 = {};
  v8f acc20 = {}, acc21 = {};
  v8f acc30 = {}, acc31 = {};

#pragma unroll 2
  for (int k = 0; k < HIDDEN; k += 4) {
    v2f a0 = *(const v2f*)(arow0 + k);
    v2f a1 = *(const v2f*)(arow1 + k);
    v2f a2 = *(const v2f*)(arow2 + k);
    v2f a3 = *(const v2f*)(arow3 + k);
    v2f b0 = *(const v2f*)(brow0 + k);
    v2f b1 = *(const v2f*)(brow1 + k);
    acc00 = __builtin_amdgcn_wmma_f32_16x16x4_f32(false, a0, false, b0, (short)0, acc00, false, false);
    acc01 = __builtin_amdgcn_wmma_f32_16x16x4_f32(false, a0, false, b1, (short)0, acc01, false, false);
    acc10 = __builtin_amdgcn_wmma_f32_16x16x4_f32(false, a1, false, b0, (short)0, acc10, false, false);
    acc11 = __builtin_amdgcn_wmma_f32_16x16x4_f32(false, a1, false, b1, (short)0, acc11, false, false);
    acc20 = __builtin_amdgcn_wmma_f32_16x16x4_f32(false, a2, false, b0, (short)0, acc20, false, false);
    acc21 = __builtin_amdgcn_wmma_f32_16x16x4_f32(false, a2, false, b1, (short)0, acc21, false, false);
    acc30 = __builtin_amdgcn_wmma_f32_16x16x4_f32(false, a3, false, b0, (short)0, acc30, false, false);
    acc31 = __builtin_amdgcn_wmma_f32_16x16x4_f32(false, a3, false, b1, (short)0, acc31, false, false);
  }

  const int row0 = (lane >> 4) << 3;          // 0 or 8
  const v8f* accs[4][2] = {{&acc00, &acc01}, {&acc10, &acc11},
                           {&acc20, &acc21}, {&acc30, &acc31}};
#pragma unroll
  for (int j = 0; j < 2; ++j) {
    const int n = tn * 32 + j * 16 + lm;
    const float bias = B1[n];
#pragma unroll
    for (int i = 0; i < 4; ++i) {
      const v8f& a = *accs[i][j];
#pragma unroll
      for (int v = 0; v < 8; ++v) {
        int mm = tm * 64 + i * 16 + row0 + v;
        float val = a[v] + bias;
        H[(size_t)mm * HIDDEN + n] = val > 0.0f ? val : 0.0f;
      }
    }
  }
}

// ---------------------------------------------------------------------------
// Kernel 3: per-token router. One wave32 per token.
//   logits[e] = h[t,:] . w2[e,:] + b2[e]; softmax; top-2; renorm;
//   write router_probs, slot-0 dispatch/combine, accumulate expert prob sums.
// ---------------------------------------------------------------------------
__global__ __launch_bounds__(32) void router_kernel(
    const float* __restrict__ H,    // [4096,1024]
    const float* __restrict__ W2,   // [8,1024]
    const float* __restrict__ B2,   // [8]
    float* __restrict__ dispatch,   // [4096,8,1536]
    float* __restrict__ combine,    // [4096,8,1536]
    float* __restrict__ probs,      // [4096,8]
    float* __restrict__ esum)       // [8]
{
  const int t = blockIdx.x;
  const int lane = threadIdx.x;
  const float* __restrict__ hrow = H + (size_t)t * HIDDEN;

  float acc[NEXP];
#pragma unroll
  for (int e = 0; e < NEXP; ++e) acc[e] = 0.0f;

  for (int k = lane; k < HIDDEN; k += 32) {
    float hv = hrow[k];
#pragma unroll
    for (int e = 0; e < NEXP; ++e) acc[e] += hv * W2[e * HIDDEN + k];
  }

  // full-wave butterfly reduction (wave32)
#pragma unroll
  for (int e = 0; e < NEXP; ++e) {
    float v = acc[e];
    for (int off = 16; off > 0; off >>= 1) v += __shfl_xor(v, off, 32);
    acc[e] = v;
  }

  if (lane == 0) {
    float logits[NEXP];
    float mx = -3.4e38f;
#pragma unroll
    for (int e = 0; e < NEXP; ++e) {
      logits[e] = acc[e] + B2[e];
      mx = fmaxf(mx, logits[e]);
    }
    float p[NEXP];
    float den = 0.0f;
#pragma unroll
    for (int e = 0; e < NEXP; ++e) { p[e] = expf(logits[e] - mx); den += p[e]; }
    float inv = 1.0f / den;
#pragma unroll
    for (int e = 0; e < NEXP; ++e) {
      p[e] *= inv;
      probs[t * NEXP + e] = p[e];
    }
    // top-2 (ties -> lowest index, matching jax.lax.top_k)
    int e1 = 0;
#pragma unroll
    for (int e = 1; e < NEXP; ++e) if (p[e] > p[e1]) e1 = e;
    int e2 = (e1 == 0) ? 1 : 0;
#pragma unroll
    for (int e = 0; e < NEXP; ++e) if (e != e1 && p[e] > p[e2]) e2 = e;

    float s = p[e1] + p[e2];
    float invs = 1.0f / s;
    dispatch[((size_t)t * NEXP + e1) * CAP] = 1.0f;
    dispatch[((size_t)t * NEXP + e2) * CAP] = 1.0f;
    combine [((size_t)t * NEXP + e1) * CAP] = p[e1] * invs;
    combine [((size_t)t * NEXP + e2) * CAP] = p[e2] * invs;

#pragma unroll
    for (int e = 0; e < NEXP; ++e) atomicAdd(&esum[e], p[e]);
  }
}

// ---------------------------------------------------------------------------
// Kernel 4: aux loss from per-expert probability sums.
// ---------------------------------------------------------------------------
__global__ void finalize_kernel(const float* __restrict__ esum,
                                float* __restrict__ out_aux) {
  if (threadIdx.x == 0 && blockIdx.x == 0) {
    float aux = 0.0f;
#pragma unroll
    for (int e = 0; e < NEXP; ++e) {
      float pm = esum[e] * (1.0f / (float)NTOK);
      aux += pm * logf(pm * (float)NEXP + 1e-9f);
    }
    *out_aux = aux;
  }
}

extern "C" void kernel_launch(void* const* d_in, const int* in_sizes, int n_in,
                              void* d_out, int out_size, void* d_ws, size_t ws_size,
                              hipStream_t stream) {
  const float* X  = (const float*)d_in[0];  // hidden_states [2,2048,1024]
  const float* W1 = (const float*)d_in[1];  // [1024,1024]
  const float* B1 = (const float*)d_in[2];  // [1024]
  const float* W2 = (const float*)d_in[3];  // [8,1024]
  const float* B2 = (const float*)d_in[4];  // [8]

  float* out  = (float*)d_out;
  float* H    = (float*)d_ws;                                   // 16 MB scratch
  float* esum = (float*)((char*)d_ws + (size_t)NTOK * HIDDEN * sizeof(float));

  // 1) zero the 402 MB dispatch/combine region + aux accumulators (NT stores)
  zero_fill_kernel<<<4096, 256, 0, stream>>>((v4f*)out, (long)PROBS_OFF / 4, esum);

  // 2) h = relu(X @ W1^T + b1) -- 2048 register-blocked WMMA waves, 4/block
  gemm16_relu_kernel<<<512, 128, 0, stream>>>(X, W1, B1, H);

  // 3) per-token routing: logits, softmax, top-2, scatter slot 0
  router_kernel<<<NTOK, 32, 0, stream>>>(H, W2, B2,
                                         out,                    // dispatch
                                         out + DISP_ELEMS,       // combine
                                         out + PROBS_OFF,        // router_probs
                                         esum);

  // 4) aux loss scalar
  finalize_kernel<<<1, 32, 0, stream>>>(esum, out + AUX_OFF);
}